// TopkLayer_60206851555927
// MI455X (gfx1250) — compile-verified
//
#include <hip/hip_runtime.h>
#include <hip/hip_bf16.h>
#include <stdint.h>

#define BATCH    64
#define SEQ      577
#define DIM      768
#define DT       32            // d-channels per workgroup tile
#define ROWF     36            // floats per LDS row: 32 data + 4 pad (144B, 16B-aligned, conflict-free)
#define KSEL     57            // max(1, int(577 * 0.1))
#define THREADS  256

typedef int v4i __attribute__((ext_vector_type(4)));
typedef __attribute__((address_space(1))) v4i gv4i;   // global 16B chunk
typedef __attribute__((address_space(3))) v4i lv4i;   // LDS 16B chunk

__device__ __forceinline__ void async_load_16B(unsigned lds_addr, unsigned long long gaddr) {
#if __has_builtin(__builtin_amdgcn_global_load_async_to_lds_b128)
    __builtin_amdgcn_global_load_async_to_lds_b128(
        (gv4i*)(uintptr_t)gaddr, (lv4i*)(uintptr_t)lds_addr, /*offset=*/0, /*cpol=*/0);
#else
    asm volatile("global_load_async_to_lds_b128 %0, %1, off"
                 :: "v"(lds_addr), "v"(gaddr) : "memory");
#endif
}

__device__ __forceinline__ void wait_async_zero() {
#if __has_builtin(__builtin_amdgcn_s_wait_asynccnt)
    __builtin_amdgcn_s_wait_asynccnt(0);
#else
    asm volatile("s_wait_asynccnt 0" ::: "memory");
#endif
}

__global__ __launch_bounds__(THREADS)
void topk_seq_kernel(const float* __restrict__ x, float* __restrict__ out) {
    __shared__ __align__(16) float tile[SEQ * ROWF];   // ~81 KB
    __shared__ unsigned thrsh[DT];

    const int tid = threadIdx.x;
    const int b   = blockIdx.y;
    const int d0  = blockIdx.x * DT;

    // ---------------- Phase 1: async-stage [SEQ x DT] tile, 16B chunks ----------------
    const unsigned lds_base =
        (unsigned)(uintptr_t)(__attribute__((address_space(3))) float*)tile;
    const unsigned long long gbase =
        (unsigned long long)(uintptr_t)x +
        ((unsigned long long)b * SEQ * DIM + (unsigned long long)d0) * 4ull;

    const int nchunks = SEQ * ((DT * 4) / 16);        // 577 * 8 chunks of 16B
    for (int t = tid; t < nchunks; t += THREADS) {
        const int s  = t >> 3;
        const int ch = t & 7;
        const unsigned            lds_addr = lds_base + (unsigned)(s * (ROWF * 4) + ch * 16);
        const unsigned long long  gaddr    = gbase + (unsigned long long)s * (DIM * 4) + ch * 16;
        async_load_16B(lds_addr, gaddr);
    }
    wait_async_zero();
    __syncthreads();

    // ---------------- Phase 2: per-channel 57th-largest |x| via bit binary search ------
    const int lane = tid & 31;
    const int wv   = tid >> 5;                        // 8 waves, 4 channels each
    for (int cc = 0; cc < DT / 8; ++cc) {
        const int c = wv * (DT / 8) + cc;

        unsigned key[19];                             // |x| bit patterns (monotone order)
        #pragma unroll
        for (int j = 0; j < 19; ++j) {
            const int s = lane + 32 * j;
            key[j] = (s < SEQ)
                   ? (__float_as_uint(tile[s * ROWF + c]) & 0x7fffffffu)
                   : 0u;                              // never >= cand (cand >= 1)
        }

        unsigned thr = 0;
        for (int bit = 30; bit >= 0; --bit) {
            const unsigned cand = thr | (1u << bit);
            int cnt = 0;
            #pragma unroll
            for (int j = 0; j < 19; ++j) cnt += (key[j] >= cand) ? 1 : 0;
            #pragma unroll
            for (int off = 16; off > 0; off >>= 1) cnt += __shfl_xor(cnt, off, 32);
            if (cnt >= KSEL) thr = cand;              // uniform across the wave
        }
        if (lane == 0) thrsh[c] = thr;
    }
    __syncthreads();

    // ---------------- Phase 3: float4 masked writeback (ds_load_b128 + b128 stores) ---
    float* outbase = out + ((size_t)b * SEQ * DIM + (size_t)d0);
    for (int t = tid; t < SEQ * 8; t += THREADS) {
        const int s  = t >> 3;
        const int ch = t & 7;                         // 16B sub-chunk within the row
        const float4 v = *(const float4*)&tile[s * ROWF + ch * 4];
        float4 r;
        r.x = ((__float_as_uint(v.x) & 0x7fffffffu) >= thrsh[ch * 4 + 0]) ? v.x : 0.0f;
        r.y = ((__float_as_uint(v.y) & 0x7fffffffu) >= thrsh[ch * 4 + 1]) ? v.y : 0.0f;
        r.z = ((__float_as_uint(v.z) & 0x7fffffffu) >= thrsh[ch * 4 + 2]) ? v.z : 0.0f;
        r.w = ((__float_as_uint(v.w) & 0x7fffffffu) >= thrsh[ch * 4 + 3]) ? v.w : 0.0f;
        *(float4*)(outbase + (size_t)s * DIM + ch * 4) = r;   // lane-contiguous 16B -> 512B/wave
    }
}

extern "C" void kernel_launch(void* const* d_in, const int* in_sizes, int n_in,
                              void* d_out, int out_size, void* d_ws, size_t ws_size,
                              hipStream_t stream) {
    (void)in_sizes; (void)n_in; (void)out_size; (void)d_ws; (void)ws_size;
    const float* x   = (const float*)d_in[0];
    float*       out = (float*)d_out;
    dim3 grid(DIM / DT, BATCH);
    topk_seq_kernel<<<grid, dim3(THREADS), 0, stream>>>(x, out);
}